// MyGNN_9225589751902
// MI455X (gfx1250) — compile-verified
//
#include <hip/hip_runtime.h>

typedef __attribute__((ext_vector_type(2))) float v2f;
typedef __attribute__((ext_vector_type(8))) float v8f;

// ------------------------------------------------------------------
// small utility kernels
// ------------------------------------------------------------------

__global__ void init_deg_pool(float* __restrict__ deg, float* __restrict__ pool, int n) {
  int i = blockIdx.x * blockDim.x + threadIdx.x;
  if (i < n) deg[i] = 1.0f;                 // self-loop contributes 1 to in-degree
  if (blockIdx.x == 0 && threadIdx.x < 32) pool[threadIdx.x] = 0.0f;
}

__global__ void deg_accum(const int* __restrict__ dst, float* __restrict__ deg, int E) {
  int i = blockIdx.x * blockDim.x + threadIdx.x;
  if (i < E) unsafeAtomicAdd(&deg[dst[i]], 1.0f);
}

__global__ void deg_to_dinv(float* __restrict__ deg, int n) {
  int i = blockIdx.x * blockDim.x + threadIdx.x;
  if (i < n) deg[i] = rsqrtf(deg[i]);       // deg >= 1 always (self loop)
}

// ------------------------------------------------------------------
// dense node-feature transform:  out[N,FOUT] = A[N,KDIM] @ W[KDIM,FOUT]
// f32 WMMA, one wave per 16-row block.
// A-frag (16x4 f32, 2 VGPR): lanes 0-15 hold K={0,1}, lanes 16-31 K={2,3}.
// B-frag (4x16 f32, 2 VGPR): mirrored (lanes 0-15 K={0,1}, 16-31 K={2,3}).
// C/D (16x16 f32, 8 VGPR): lane&15 = col, lanes>=16 own rows M+8.
// K-tail: always load from clamped in-bounds address, select 0.0f via
// v_cndmask (no exec-mask branches between WMMAs).
// Store epilogue: wave-uniform full-tile fast path -> unmasked clause-able
// stores; only the final partial tile takes the per-row masked path.
// ------------------------------------------------------------------
template <int KDIM, int FOUT>
__global__ void gemm_wmma_f32(const float* __restrict__ A, const float* __restrict__ W,
                              float* __restrict__ out, int nrows) {
  const int lane = threadIdx.x & 31;
  const int wave = threadIdx.x >> 5;
  const int wavesPerBlock = blockDim.x >> 5;
  const int blockRow = (blockIdx.x * wavesPerBlock + wave) * 16;
  if (blockRow >= nrows) return;            // wave-uniform: EXEC all-1s below
  const int half = lane >> 4;               // 0 -> K pair {0,1}; 1 -> K pair {2,3}
  const int l16  = lane & 15;

  constexpr int NT = FOUT / 16;             // 16-column tiles
  constexpr int KC = (KDIM + 3) / 4;        // K chunks of 4 (zero padded)

  v8f c[NT] = {};

  int mrow = blockRow + l16;
  if (mrow >= nrows) mrow = nrows - 1;      // clamp loads; stores masked later

#pragma unroll
  for (int kc = 0; kc < KC; ++kc) {
    const int k0 = kc * 4 + half * 2;
    const bool in0 = (k0     < KDIM);
    const bool in1 = (k0 + 1 < KDIM);
    const int ka0 = in0 ? k0     : 0;       // clamped, always in-bounds
    const int ka1 = in1 ? k0 + 1 : 0;
    v2f a;
    {
      float va0 = A[(long long)mrow * KDIM + ka0];
      float va1 = A[(long long)mrow * KDIM + ka1];
      a.x = in0 ? va0 : 0.0f;
      a.y = in1 ? va1 : 0.0f;
    }
#pragma unroll
    for (int t = 0; t < NT; ++t) {
      const int col = t * 16 + l16;
      v2f b;
      float vb0 = W[ka0 * FOUT + col];
      float vb1 = W[ka1 * FOUT + col];
      b.x = in0 ? vb0 : 0.0f;
      b.y = in1 ? vb1 : 0.0f;
      c[t] = __builtin_amdgcn_wmma_f32_16x16x4_f32(false, a, false, b, (short)0,
                                                   c[t], false, false);
    }
  }

  // ---- store epilogue ----
  // Base pointer for this lane: row (blockRow + half*8), column l16.
  float* obase = out + (long long)(blockRow + half * 8) * FOUT + l16;
  if (blockRow + 16 <= nrows) {
    // Full tile (wave-uniform): straight-line stores, immediate offsets.
#pragma unroll
    for (int t = 0; t < NT; ++t)
#pragma unroll
      for (int r = 0; r < 8; ++r)
        obase[r * FOUT + t * 16] = c[t][r];
  } else {
    // Partial tail tile: per-row mask.
#pragma unroll
    for (int t = 0; t < NT; ++t)
#pragma unroll
      for (int r = 0; r < 8; ++r)
        if (blockRow + half * 8 + r < nrows) obase[r * FOUT + t * 16] = c[t][r];
  }
}

// ------------------------------------------------------------------
// aggregation kernels
// ------------------------------------------------------------------

// agg[i,:] = h[i,:] * dinv[i]^2   (self-loop contribution), float4-wide
template <int F>
__global__ void selfloop_scale(const float* __restrict__ h, const float* __restrict__ dinv,
                               float* __restrict__ agg, int n) {
  constexpr int G = F / 4;
  int idx = blockIdx.x * blockDim.x + threadIdx.x;
  if (idx >= n * G) return;
  int row = idx / G;
  float dv = dinv[row];
  dv *= dv;
  const float4 hv = *reinterpret_cast<const float4*>(h + (long long)idx * 4);
  float4 o;
  o.x = hv.x * dv; o.y = hv.y * dv; o.z = hv.z * dv; o.w = hv.w * dv;
  *reinterpret_cast<float4*>(agg + (long long)idx * 4) = o;
}

// per edge: agg[dst,:] += h[src,:] * dinv[src]*dinv[dst]; F/4 threads per edge
template <int F>
__global__ void edge_scatter(const int* __restrict__ src, const int* __restrict__ dst,
                             const float* __restrict__ h, const float* __restrict__ dinv,
                             float* __restrict__ agg, int E) {
  constexpr int G = F / 4;
  int idx = blockIdx.x * blockDim.x + threadIdx.x;
  if (idx >= E * G) return;
  int e  = idx / G;
  int g4 = (idx - e * G) * 4;
  int s = src[e];
  int d = dst[e];
  float norm = dinv[s] * dinv[d];
  const float4 hv = *reinterpret_cast<const float4*>(h + (long long)s * F + g4);
  float* base = agg + (long long)d * F + g4;
  unsafeAtomicAdd(base + 0, hv.x * norm);
  unsafeAtomicAdd(base + 1, hv.y * norm);
  unsafeAtomicAdd(base + 2, hv.z * norm);
  unsafeAtomicAdd(base + 3, hv.w * norm);
}

template <int F>
__global__ void bias_relu(float* __restrict__ agg, const float* __restrict__ b, int n) {
  int idx = blockIdx.x * blockDim.x + threadIdx.x;
  if (idx >= n * F) return;
  int f = idx & (F - 1);
  float v = agg[idx] + b[f];
  agg[idx] = v > 0.0f ? v : 0.0f;
}

// ------------------------------------------------------------------
// mean pool (32 features) + final linear
// ------------------------------------------------------------------

__global__ void pool_kernel(const float* __restrict__ act, float* __restrict__ pool, int n) {
  __shared__ float sdata[256];
  const int f    = threadIdx.x & 31;
  const int wave = threadIdx.x >> 5;
  const int wavesPerBlock = blockDim.x >> 5;
  int row    = blockIdx.x * wavesPerBlock + wave;
  int stride = gridDim.x * wavesPerBlock;
  float s = 0.0f;
  for (; row < n; row += stride) s += act[(long long)row * 32 + f];
  sdata[threadIdx.x] = s;
  __syncthreads();
  if (threadIdx.x < 32) {
    float tot = 0.0f;
    for (int w = 0; w < wavesPerBlock; ++w) tot += sdata[w * 32 + f];
    unsafeAtomicAdd(&pool[f], tot);
  }
}

__global__ void final_kernel(const float* __restrict__ pool, const float* __restrict__ Wo,
                             const float* __restrict__ bo, float* __restrict__ out, int n) {
  if (threadIdx.x == 0 && blockIdx.x == 0) {
    const float inv_n = 1.0f / (float)n;
    float acc = 0.0f;
    for (int f = 0; f < 32; ++f) acc += (pool[f] * inv_n) * Wo[f];
    out[0] = acc + bo[0];
  }
}

// ------------------------------------------------------------------
// launcher
// ------------------------------------------------------------------

extern "C" void kernel_launch(void* const* d_in, const int* in_sizes, int n_in,
                              void* d_out, int out_size, void* d_ws, size_t ws_size,
                              hipStream_t stream) {
  const float* x   = (const float*)d_in[0];
  const int*   ei  = (const int*)  d_in[1];
  const float* W1  = (const float*)d_in[2];
  const float* b1  = (const float*)d_in[3];
  const float* W2  = (const float*)d_in[4];
  const float* b2  = (const float*)d_in[5];
  const float* Wo  = (const float*)d_in[6];
  const float* bo  = (const float*)d_in[7];
  float*       out = (float*)d_out;

  const int N = in_sizes[0] / 9;
  const int E = in_sizes[1] / 2;
  const int* srcp = ei;
  const int* dstp = ei + E;

  // workspace layout (floats), 16-float aligned slabs
  float* ws = (float*)d_ws;
  size_t off = 0;
  auto alloc = [&](size_t nf) { float* p = ws + off; off += (nf + 15) & ~(size_t)15; return p; };
  float* dinv = alloc((size_t)N);        // rsqrt degree
  float* bufA = alloc((size_t)N * 64);   // h1, later h2 (first 32N)
  float* bufB = alloc((size_t)N * 64);   // agg1/act1, later agg2/act2 (first 32N)
  float* pool = alloc(32);

  const int TB = 256;
  auto nb = [&](long long work) { return (unsigned)((work + TB - 1) / TB); };
  const unsigned gemmBlocks = (unsigned)((((N + 15) / 16) + 7) / 8); // 8 waves/block

  // degrees -> dinv
  init_deg_pool<<<nb(N), TB, 0, stream>>>(dinv, pool, N);
  deg_accum<<<nb(E), TB, 0, stream>>>(dstp, dinv, E);
  deg_to_dinv<<<nb(N), TB, 0, stream>>>(dinv, N);

  // ---- layer 1: h1 = x @ W1 ; agg ; +b1 ; relu ----
  gemm_wmma_f32<9, 64><<<gemmBlocks, TB, 0, stream>>>(x, W1, bufA, N);
  selfloop_scale<64><<<nb((long long)N * 16), TB, 0, stream>>>(bufA, dinv, bufB, N);
  edge_scatter<64><<<nb((long long)E * 16), TB, 0, stream>>>(srcp, dstp, bufA, dinv, bufB, E);
  bias_relu<64><<<nb((long long)N * 64), TB, 0, stream>>>(bufB, b1, N);

  // ---- layer 2: h2 = act1 @ W2 ; agg ; +b2 ; relu ----
  gemm_wmma_f32<64, 32><<<gemmBlocks, TB, 0, stream>>>(bufB, W2, bufA, N);   // h2 -> bufA[0..32N)
  selfloop_scale<32><<<nb((long long)N * 8), TB, 0, stream>>>(bufA, dinv, bufB, N); // agg2 -> bufB
  edge_scatter<32><<<nb((long long)E * 8), TB, 0, stream>>>(srcp, dstp, bufA, dinv, bufB, E);
  bias_relu<32><<<nb((long long)N * 32), TB, 0, stream>>>(bufB, b2, N);

  // ---- mean pool + linear head ----
  pool_kernel<<<256, TB, 0, stream>>>(bufB, pool, N);
  final_kernel<<<1, 32, 0, stream>>>(pool, Wo, bo, out, N);
}